// E8ActionReducer_43224550867302
// MI455X (gfx1250) — compile-verified
//
#include <hip/hip_runtime.h>
#include <cmath>

typedef __attribute__((ext_vector_type(2))) float v2f;
typedef __attribute__((ext_vector_type(4))) float v4f;
typedef __attribute__((ext_vector_type(8))) float v8f;

// ---------------- E8 root table (240 roots, reference ordering) -------------
struct RootTable { float v[240][8]; };

static constexpr RootTable make_roots() {
  RootTable t{};
  int n = 0;
  // Type 1: +-1 at positions i<j, sign order (+,+),(+,-),(-,+),(-,-)
  for (int i = 0; i < 8; ++i)
    for (int j = i + 1; j < 8; ++j)
      for (int si = 0; si < 2; ++si)
        for (int sj = 0; sj < 2; ++sj) {
          for (int c = 0; c < 8; ++c) t.v[n][c] = 0.0f;
          t.v[n][i] = si ? -1.0f : 1.0f;
          t.v[n][j] = sj ? -1.0f : 1.0f;
          ++n;
        }
  // Type 2: all +-0.5 with even number of minus signs, itertools.product order
  for (int m = 0; m < 256; ++m) {
    int neg = 0;
    for (int bpos = 0; bpos < 8; ++bpos) neg += (m >> bpos) & 1;
    if ((neg & 1) == 0) {
      for (int c = 0; c < 8; ++c)
        t.v[n][c] = ((m >> (7 - c)) & 1) ? -0.5f : 0.5f;
      ++n;
    }
  }
  return t;
}

__constant__ RootTable g_roots = make_roots();

// ---------------------------------------------------------------------------
__global__ __launch_bounds__(256)
void e8_reducer_kernel(
    const float* __restrict__ co, const float* __restrict__ conf,
    const float* __restrict__ Wq, const float* __restrict__ bq,
    const float* __restrict__ Wk, const float* __restrict__ bk,
    const float* __restrict__ Wv, const float* __restrict__ bv,
    const float* __restrict__ comm, const float* __restrict__ role,
    const float* __restrict__ mha_w, const float* __restrict__ mha_b,
    const float* __restrict__ W1, const float* __restrict__ b1,
    const float* __restrict__ W2, const float* __restrict__ b2,
    const float* __restrict__ gate,
    float* __restrict__ outQ, int* __restrict__ outI, float* __restrict__ outW,
    int Btot)
{
  __shared__ float sWq[256], sWk[256], sW1[256], sW2[256];
  __shared__ float sbq[32], sbk[32], sb1[32];
  __shared__ float sWv[64], sWq3[64], sWk3[64];
  __shared__ float sbv[8], sb2[8], sbq3[8], sbk3[8];
  __shared__ float sCsm[49], sRole[7];
  __shared__ float sGate;
  __shared__ float sRef[256 * 9];    // padded stride 9 to spread LDS banks
  __shared__ float sRoots[240 * 9];  // root table staged to LDS, stride 9

  const int tid = threadIdx.x;

  // ---- stage block-uniform parameters + root table into LDS ----
  sWq[tid] = Wq[tid]; sWk[tid] = Wk[tid]; sW1[tid] = W1[tid]; sW2[tid] = W2[tid];
  if (tid < 32) { sbq[tid] = bq[tid]; sbk[tid] = bk[tid]; sb1[tid] = b1[tid]; }
  if (tid < 64) { sWv[tid] = Wv[tid]; sWq3[tid] = mha_w[tid]; sWk3[tid] = mha_w[64 + tid]; }
  if (tid < 8)  { sbv[tid] = bv[tid]; sb2[tid] = b2[tid];
                  sbq3[tid] = mha_b[tid]; sbk3[tid] = mha_b[8 + tid]; }
  {
    const float* rsrc = &g_roots.v[0][0];
    #pragma unroll
    for (int i = tid; i < 1920; i += 256)          // 240*8 elements
      sRoots[(i >> 3) * 9 + (i & 7)] = rsrc[i];
  }
  if (tid < 7) {                         // softmax(comm) row tid
    float r[7], mx = -3.0e38f;
    #pragma unroll
    for (int m = 0; m < 7; ++m) { r[m] = comm[tid * 7 + m]; mx = fmaxf(mx, r[m]); }
    float s = 0.0f;
    #pragma unroll
    for (int m = 0; m < 7; ++m) { r[m] = __expf(r[m] - mx); s += r[m]; }
    float is = 1.0f / s;
    #pragma unroll
    for (int m = 0; m < 7; ++m) sCsm[tid * 7 + m] = r[m] * is;
  } else if (tid == 7) {                 // softmax(role)
    float r[7], mx = -3.0e38f;
    #pragma unroll
    for (int m = 0; m < 7; ++m) { r[m] = role[m]; mx = fmaxf(mx, r[m]); }
    float s = 0.0f;
    #pragma unroll
    for (int m = 0; m < 7; ++m) { r[m] = __expf(r[m] - mx); s += r[m]; }
    float is = 1.0f / s;
    #pragma unroll
    for (int m = 0; m < 7; ++m) sRole[m] = r[m] * is;
  } else if (tid == 8) {
    sGate = 1.0f / (1.0f + __expf(-gate[0]));
  }
  __syncthreads();

  const int gb = blockIdx.x * 256 + tid;
  const int b  = (gb < Btot) ? gb : (Btot - 1);

  // ---- load (non-temporal: streamed once) + normalize x (7x8) ----
  float x[56];
  {
    const v4f* xg = (const v4f*)(co + (size_t)b * 56);
    #pragma unroll
    for (int i = 0; i < 14; ++i) {
      v4f t4 = __builtin_nontemporal_load(xg + i);
      x[i*4+0] = t4[0]; x[i*4+1] = t4[1]; x[i*4+2] = t4[2]; x[i*4+3] = t4[3];
    }
  }
  #pragma unroll
  for (int n = 0; n < 7; ++n) {
    float ss = 0.0f;
    #pragma unroll
    for (int c = 0; c < 8; ++c) { float xv = x[n*8+c]; ss += xv * xv; }
    float inv = 1.0f / fmaxf(sqrtf(ss), 1e-12f);
    #pragma unroll
    for (int c = 0; c < 8; ++c) x[n*8+c] *= inv;
  }

  // ---- scores path 1: q = Wq@mean(x)+bq ; score_n = (Wk^T q)·x_n + q·bk ----
  float xm[8];
  #pragma unroll
  for (int c = 0; c < 8; ++c) {
    float s = 0.0f;
    #pragma unroll
    for (int n = 0; n < 7; ++n) s += x[n*8+c];
    xm[c] = s * (1.0f / 7.0f);
  }
  float u[8] = {0,0,0,0,0,0,0,0};
  float qbk = 0.0f;
  #pragma unroll
  for (int h = 0; h < 32; ++h) {
    float qh = sbq[h];
    #pragma unroll
    for (int c = 0; c < 8; ++c) qh += sWq[h*8+c] * xm[c];
    #pragma unroll
    for (int c = 0; c < 8; ++c) u[c] += sWk[h*8+c] * qh;
    qbk += qh * sbk[h];
  }
  float s1[7];
  {
    float s0[7];
    #pragma unroll
    for (int n = 0; n < 7; ++n) {
      float d = qbk;
      #pragma unroll
      for (int c = 0; c < 8; ++c) d += u[c] * x[n*8+c];
      s0[n] = d * 0.1767766952966369f;           // 1/sqrt(32)
    }
    #pragma unroll
    for (int m = 0; m < 7; ++m) {
      float d = sRole[m];
      #pragma unroll
      for (int n = 0; n < 7; ++n) d += s0[n] * sCsm[n*7+m];
      s1[m] = d;
    }
  }

  // ---- mini-MHA head-mean scores ----
  float K3[56];
  #pragma unroll
  for (int n = 0; n < 7; ++n) {
    #pragma unroll
    for (int hd = 0; hd < 8; ++hd) {
      float a = sbk3[hd];
      #pragma unroll
      for (int c = 0; c < 8; ++c) a += sWk3[hd*8+c] * x[n*8+c];
      K3[n*8+hd] = a;
    }
  }
  float mhs[7] = {0,0,0,0,0,0,0};
  #pragma unroll
  for (int n = 0; n < 7; ++n) {
    float Qn[8];
    #pragma unroll
    for (int hd = 0; hd < 8; ++hd) {
      float a = sbq3[hd];
      #pragma unroll
      for (int c = 0; c < 8; ++c) a += sWq3[hd*8+c] * x[n*8+c];
      Qn[hd] = a;
    }
    #pragma unroll
    for (int h = 0; h < 4; ++h) {
      float lg[7], mx = -3.0e38f;
      #pragma unroll
      for (int m = 0; m < 7; ++m) {
        lg[m] = (Qn[2*h] * K3[m*8+2*h] + Qn[2*h+1] * K3[m*8+2*h+1])
                * 0.7071067811865475f;           // 1/sqrt(2)
        mx = fmaxf(mx, lg[m]);
      }
      float s = 0.0f;
      #pragma unroll
      for (int m = 0; m < 7; ++m) { lg[m] = __expf(lg[m] - mx); s += lg[m]; }
      float is = 1.0f / s;
      #pragma unroll
      for (int m = 0; m < 7; ++m) mhs[m] += lg[m] * is;
    }
  }

  // ---- weights ----
  float w[7];
  {
    float mx = -3.0e38f;
    #pragma unroll
    for (int m = 0; m < 7; ++m) {
      w[m] = 0.7f * s1[m] + 0.3f * (mhs[m] * (1.0f / 28.0f));
      mx = fmaxf(mx, w[m]);
    }
    float s = 0.0f;
    #pragma unroll
    for (int m = 0; m < 7; ++m) { w[m] = __expf(w[m] - mx); s += w[m]; }
    float is = 1.0f / s, sw = 0.0f;
    #pragma unroll
    for (int m = 0; m < 7; ++m) { w[m] = w[m] * is * conf[(size_t)b*7 + m]; sw += w[m]; }
    float isw = 1.0f / fmaxf(sw, 1e-8f);
    #pragma unroll
    for (int m = 0; m < 7; ++m) w[m] *= isw;
  }

  // ---- combined = sum_n w_n (Wv x_n + bv) = Wv (sum w_n x_n) + bv*sum(w) ----
  float xw[8], Wsum = 0.0f;
  #pragma unroll
  for (int m = 0; m < 7; ++m) Wsum += w[m];
  #pragma unroll
  for (int c = 0; c < 8; ++c) {
    float a = 0.0f;
    #pragma unroll
    for (int n = 0; n < 7; ++n) a += w[n] * x[n*8+c];
    xw[c] = a;
  }
  float comb[8];
  #pragma unroll
  for (int c = 0; c < 8; ++c) {
    float a = sbv[c] * Wsum;
    #pragma unroll
    for (int d0 = 0; d0 < 8; ++d0) a += sWv[c*8+d0] * xw[d0];
    comb[c] = a;
  }

  // ---- MLP (exact gelu) + gated residual + normalize ----
  float out8[8];
  #pragma unroll
  for (int c = 0; c < 8; ++c) out8[c] = sb2[c];
  #pragma unroll
  for (int j = 0; j < 32; ++j) {
    float hj = sb1[j];
    #pragma unroll
    for (int c = 0; c < 8; ++c) hj += sW1[j*8+c] * comb[c];
    float gj = 0.5f * hj * (1.0f + erff(hj * 0.7071067811865475f));
    #pragma unroll
    for (int c = 0; c < 8; ++c) out8[c] += sW2[c*32+j] * gj;
  }
  float refined[8];
  {
    float gg = sGate, nn = 0.0f;
    #pragma unroll
    for (int c = 0; c < 8; ++c) { refined[c] = comb[c] + gg * out8[c]; nn += refined[c]*refined[c]; }
    float inv = 1.0f / fmaxf(sqrtf(nn), 1e-12f);
    #pragma unroll
    for (int c = 0; c < 8; ++c) refined[c] *= inv;
  }

  // ---- exact E8 hard quantization (no dynamic private indexing) ----
  float y0[8], y1[8];
  {
    float f[8], sum = 0.0f, ba = -1.0f, be = 0.0f; int bi = 0;
    #pragma unroll
    for (int c = 0; c < 8; ++c) {
      f[c] = rintf(refined[c]);
      float e = refined[c] - f[c];
      sum += f[c];
      float a = fabsf(e);
      if (a > ba) { ba = a; bi = c; be = e; }
    }
    bool even = (fmodf(sum, 2.0f) == 0.0f);
    float adj = even ? 0.0f : ((be >= 0.0f) ? 1.0f : -1.0f);
    #pragma unroll
    for (int c = 0; c < 8; ++c) y0[c] = f[c] + ((c == bi) ? adj : 0.0f);
  }
  {
    float f[8], sum = 0.0f, ba = -1.0f, be = 0.0f; int bi = 0;
    #pragma unroll
    for (int c = 0; c < 8; ++c) {
      float xc = refined[c] - 0.5f;
      f[c] = rintf(xc);
      float e = xc - f[c];
      sum += f[c];
      float a = fabsf(e);
      if (a > ba) { ba = a; bi = c; be = e; }
    }
    bool even = (fmodf(sum, 2.0f) == 0.0f);
    float adj = even ? 0.0f : ((be >= 0.0f) ? 1.0f : -1.0f);
    #pragma unroll
    for (int c = 0; c < 8; ++c) y1[c] = f[c] + ((c == bi) ? adj : 0.0f) + 0.5f;
  }
  float d0s = 0.0f, d1s = 0.0f;
  #pragma unroll
  for (int c = 0; c < 8; ++c) {
    float e0 = refined[c] - y0[c]; d0s += e0 * e0;
    float e1 = refined[c] - y1[c]; d1s += e1 * e1;
  }
  const bool take0 = (d0s <= d1s);

  if (gb < Btot) {
    v4f q0, q1;
    q0[0] = take0 ? y0[0] : y1[0]; q0[1] = take0 ? y0[1] : y1[1];
    q0[2] = take0 ? y0[2] : y1[2]; q0[3] = take0 ? y0[3] : y1[3];
    q1[0] = take0 ? y0[4] : y1[4]; q1[1] = take0 ? y0[5] : y1[5];
    q1[2] = take0 ? y0[6] : y1[6]; q1[3] = take0 ? y0[7] : y1[7];
    v4f* qp = (v4f*)(outQ + (size_t)gb * 8);
    __builtin_nontemporal_store(q0, qp);
    __builtin_nontemporal_store(q1, qp + 1);
    #pragma unroll
    for (int m = 0; m < 7; ++m)
      __builtin_nontemporal_store(w[m], outW + (size_t)gb*7 + m);
  }

  // ---- WMMA codebook argmax: all roots have |root|^2==2 -> argmax(r·root) ----
  #pragma unroll
  for (int c = 0; c < 8; ++c) sRef[tid*9 + c] = refined[c];
  __syncthreads();

  const int lane = tid & 31;
  const int wv   = tid >> 5;
  const int ln   = lane & 15;
  const int kb   = (lane < 16) ? 0 : 2;      // K-pair handled by this lane half
  const int ioff = (lane < 16) ? 0 : 8;      // root-index offset of this half

  // B operands (refined^T) for both 16-row groups of this wave, loaded once
  const float* rp0 = &sRef[(wv*32 + ln) * 9];
  const float* rp1 = &sRef[(wv*32 + 16 + ln) * 9];
  v2f b0Lo = { rp0[kb],     rp0[kb + 1] };
  v2f b0Hi = { rp0[kb + 4], rp0[kb + 5] };
  v2f b1Lo = { rp1[kb],     rp1[kb + 1] };
  v2f b1Hi = { rp1[kb + 4], rp1[kb + 5] };

  float best0 = -3.0e38f, best1 = -3.0e38f;
  int   bidx0 = 0,        bidx1 = 0;

  #pragma unroll 1
  for (int t = 0; t < 15; ++t) {             // 15 tiles of 16 roots
    const float* ra = &sRoots[(t*16 + ln) * 9];  // A operand shared by both groups
    v2f aLo = { ra[kb],     ra[kb + 1] };
    v2f aHi = { ra[kb + 4], ra[kb + 5] };
    v8f d0 = {0.0f, 0.0f, 0.0f, 0.0f, 0.0f, 0.0f, 0.0f, 0.0f};
    v8f d1 = {0.0f, 0.0f, 0.0f, 0.0f, 0.0f, 0.0f, 0.0f, 0.0f};
    d0 = __builtin_amdgcn_wmma_f32_16x16x4_f32(false, aLo, false, b0Lo,
                                               (short)0, d0, false, false);
    d0 = __builtin_amdgcn_wmma_f32_16x16x4_f32(false, aHi, false, b0Hi,
                                               (short)0, d0, false, false);
    d1 = __builtin_amdgcn_wmma_f32_16x16x4_f32(false, aLo, false, b1Lo,
                                               (short)0, d1, false, false);
    d1 = __builtin_amdgcn_wmma_f32_16x16x4_f32(false, aHi, false, b1Hi,
                                               (short)0, d1, false, false);
    // lane holds row (group base + ln): d[r] = dot(row, root t*16 + r + ioff)
    const int ib = t*16 + ioff;
    #pragma unroll
    for (int r = 0; r < 8; ++r) {
      float v0 = d0[r];
      if (v0 > best0) { best0 = v0; bidx0 = ib + r; }
      float v1 = d1[r];
      if (v1 > best1) { best1 = v1; bidx1 = ib + r; }
    }
  }

  // merge the two lane halves (roots +0..7 vs +8..15), keep smallest index on tie
  {
    float ov = __shfl_xor(best0, 16, 32);
    int   oi = __shfl_xor(bidx0, 16, 32);
    if (ov > best0 || (ov == best0 && oi < bidx0)) { best0 = ov; bidx0 = oi; }
    const int row = blockIdx.x * 256 + wv*32 + ln;
    if (lane < 16 && row < Btot) __builtin_nontemporal_store(bidx0, outI + row);
  }
  {
    float ov = __shfl_xor(best1, 16, 32);
    int   oi = __shfl_xor(bidx1, 16, 32);
    if (ov > best1 || (ov == best1 && oi < bidx1)) { best1 = ov; bidx1 = oi; }
    const int row = blockIdx.x * 256 + wv*32 + 16 + ln;
    if (lane < 16 && row < Btot) __builtin_nontemporal_store(bidx1, outI + row);
  }
}

// ---------------------------------------------------------------------------
extern "C" void kernel_launch(void* const* d_in, const int* in_sizes, int n_in,
                              void* d_out, int out_size, void* d_ws, size_t ws_size,
                              hipStream_t stream) {
  (void)n_in; (void)out_size; (void)d_ws; (void)ws_size;
  const int Btot = in_sizes[0] / 56;                 // (B, 7, 8) f32
  float* outQ = (float*)d_out;                       // quantized: B*8 f32
  int*   outI = (int*)((float*)d_out + (size_t)Btot * 8);   // indices: B i32
  float* outW = (float*)d_out + (size_t)Btot * 9;    // weights: B*7 f32
  dim3 grid((Btot + 255) / 256), block(256);
  hipLaunchKernelGGL(e8_reducer_kernel, grid, block, 0, stream,
      (const float*)d_in[0],  (const float*)d_in[1],  (const float*)d_in[2],
      (const float*)d_in[3],  (const float*)d_in[4],  (const float*)d_in[5],
      (const float*)d_in[6],  (const float*)d_in[7],  (const float*)d_in[8],
      (const float*)d_in[9],  (const float*)d_in[10], (const float*)d_in[11],
      (const float*)d_in[12], (const float*)d_in[13], (const float*)d_in[14],
      (const float*)d_in[15], (const float*)d_in[16],
      outQ, outI, outW, Btot);
}